// GradientVarianceLoss_60473139527870
// MI455X (gfx1250) — compile-verified
//
#include <hip/hip_runtime.h>
#include <stdint.h>

#define Wd   1024
#define Hd   1024
#define TILE 64

typedef __attribute__((ext_vector_type(2))) float v2f;
typedef __attribute__((ext_vector_type(8))) float v8f;

// ---------------------------------------------------------------------------
// CDNA5 f32 WMMA: D(16x16) = A(16x4) * B(4x16) + C   (v_wmma_f32_16x16x4_f32)
// A layout: lanes 0-15 hold M=0..15 K={0,1}; lanes 16-31 hold M=0..15 K={2,3}
// C/D layout: VGPR r = row (r + 8*(lane>=16)), N = lane%16
// ---------------------------------------------------------------------------
__device__ __forceinline__ v8f wmma16x16x4(v2f a, v2f b, v8f c) {
  return __builtin_amdgcn_wmma_f32_16x16x4_f32(
      /*neg_a=*/false, a, /*neg_b=*/false, b,
      /*c_mod=*/(short)0, c, /*reuse_a=*/false, /*reuse_b=*/false);
}

// async 16B global -> LDS copy per lane (ASYNCcnt tracked, no dest VGPRs)
__device__ __forceinline__ void async_ld16(uint32_t lds_addr, uint64_t gaddr) {
  asm volatile("global_load_async_to_lds_b128 %0, %1, off"
               :: "v"(lds_addr), "v"(gaddr) : "memory");
}
__device__ __forceinline__ void wait_async0() {
  asm volatile("s_wait_asynccnt 0" ::: "memory");
}

__global__ void gvl_zero(float* loss) {
  if (threadIdx.x == 0) loss[0] = 0.f;
}

// One workgroup = 64x64 gradient tile of one batch image pair.
// LDS stages a 66x66 halo tile for both images (stride 72, +3 col shift so
// every async B128 write is 16B aligned; boundary chunks land in padding).
__global__ __launch_bounds__(256) void gvl_kernel(const float* __restrict__ outp,
                                                  const float* __restrict__ tgtp,
                                                  float* __restrict__ loss) {
  __shared__ __attribute__((aligned(16))) float sh[2][66][72];
  __shared__ float red[8];

  const int tid  = threadIdx.x;
  const int lane = tid & 31;
  const int wid  = tid >> 5;
  const int x0   = blockIdx.x * TILE;
  const int y0   = blockIdx.y * TILE;
  const size_t ibase = (size_t)blockIdx.z * (size_t)(Wd * Hd);
  const float* srcs[2] = { outp + ibase, tgtp + ibase };

  // phase 1: zero the halo frame (covers image-border zero padding where the
  // async loads below are predicated off)
  for (int i = tid; i < 2 * 66; i += 256) {
    int img = i / 66, r = i % 66;
    sh[img][r][0 + 3]  = 0.f;   // cc = 0   (left halo col)
    sh[img][r][65 + 3] = 0.f;   // cc = 65  (right halo col)
  }
  for (int i = tid; i < 2 * 72; i += 256) {
    int img = i / 72, c = i % 72;
    sh[img][0][c]  = 0.f;       // top halo row
    sh[img][65][c] = 0.f;       // bottom halo row
  }
  __syncthreads();

  // phase 2: async global->LDS staging.  staged cell (r, cc): gy = y0-1+r,
  // gx = x0-1+cc, stored at sh[img][r][cc+3].  Chunk c covers gx0 = x0-4+4c.
  for (int task = tid; task < 66 * 18; task += 256) {
    int r   = task / 18;
    int c   = task % 18;
    int gy  = y0 - 1 + r;
    int gx0 = x0 - 4 + 4 * c;                    // 16B-aligned column base
    bool ok = (gy >= 0) && (gy < Hd) && (gx0 >= 0) && (gx0 + 3 < Wd);
    if (ok) {
#pragma unroll
      for (int img = 0; img < 2; ++img) {
        uint32_t lds = (uint32_t)(uintptr_t)&sh[img][r][4 * c];
        uint64_t ga  = (uint64_t)(uintptr_t)(srcs[img] + (size_t)gy * Wd + gx0);
        async_ld16(lds, ga);
      }
    }
  }
  wait_async0();
  __syncthreads();

  // phase 3: Sobel + 4x4 block variance via f32 WMMA pooling.
  // Wave handles 16x16 sub-tiles; lane->pixel mapping matches WMMA A layout.
  const float inv16 = 0.0625f;
  float lacc = 0.f;
  const int m  = lane & 15;          // matrix row
  const int kb = (lane >> 4) << 1;   // K base within 4-col chunk

  for (int t = wid; t < 16; t += 8) {
    const int sy = (t >> 2) * 16;
    const int sx = (t & 3) * 16;
    float vprev[4];
#pragma unroll
    for (int img = 0; img < 2; ++img) {
      v8f csx = {}, cqx = {}, csy = {}, cqy = {};
#pragma unroll
      for (int c = 0; c < 4; ++c) {
        const int ti = sy + m;                 // gradient row in tile
        const int tj = sx + 4 * c + kb;        // gradient col of first pixel
        float a0[4], a1[4], a2[4];
#pragma unroll
        for (int dc = 0; dc < 4; ++dc) {       // input window rows ti..ti+2
          a0[dc] = sh[img][ti + 0][tj + dc + 3];
          a1[dc] = sh[img][ti + 1][tj + dc + 3];
          a2[dc] = sh[img][ti + 2][tj + dc + 3];
        }
        v2f gx, gy;
        gx.x = (a0[2] - a0[0]) + 2.f * (a1[2] - a1[0]) + (a2[2] - a2[0]);
        gx.y = (a0[3] - a0[1]) + 2.f * (a1[3] - a1[1]) + (a2[3] - a2[1]);
        gy.x = (a2[0] + 2.f * a2[1] + a2[2]) - (a0[0] + 2.f * a0[1] + a0[2]);
        gy.y = (a2[1] + 2.f * a2[2] + a2[3]) - (a0[1] + 2.f * a0[2] + a0[3]);
        // B_c: ones in column c -> D col n<4 = horizontal 4-sum of block n
        float bv = ((lane & 15) == c) ? 1.f : 0.f;
        v2f b; b.x = bv; b.y = bv;
        csx = wmma16x16x4(gx, b, csx);
        csy = wmma16x16x4(gy, b, csy);
        v2f gx2 = gx * gx, gy2 = gy * gy;
        cqx = wmma16x16x4(gx2, b, cqx);
        cqy = wmma16x16x4(gy2, b, cqy);
      }
      // vertical pooling: rows {0-3}/{4-7} per half-wave -> 2 block sums/lane
      float sxa = csx[0]+csx[1]+csx[2]+csx[3], sxb = csx[4]+csx[5]+csx[6]+csx[7];
      float qxa = cqx[0]+cqx[1]+cqx[2]+cqx[3], qxb = cqx[4]+cqx[5]+cqx[6]+cqx[7];
      float sya = csy[0]+csy[1]+csy[2]+csy[3], syb = csy[4]+csy[5]+csy[6]+csy[7];
      float qya = cqy[0]+cqy[1]+cqy[2]+cqy[3], qyb = cqy[4]+cqy[5]+cqy[6]+cqy[7];
      float mxa = sxa * inv16, mxb = sxb * inv16;
      float mya = sya * inv16, myb = syb * inv16;
      float v0 = qxa * inv16 - mxa * mxa;
      float v1 = qxb * inv16 - mxb * mxb;
      float v2 = qya * inv16 - mya * mya;
      float v3 = qyb * inv16 - myb * myb;
      if (img == 0) {
        vprev[0] = v0; vprev[1] = v1; vprev[2] = v2; vprev[3] = v3;
      } else {
        float d0 = vprev[0] - v0, d1 = vprev[1] - v1;
        float d2 = vprev[2] - v2, d3 = vprev[3] - v3;
        // lanes with D-column >= 4 carry exact zeros for both images -> add 0
        lacc += d0 * d0 + d1 * d1 + d2 * d2 + d3 * d3;
      }
    }
  }

  // phase 4: wave32 tree reduction -> LDS -> one fp32 atomic per workgroup
#pragma unroll
  for (int off = 16; off > 0; off >>= 1)
    lacc += __shfl_down(lacc, off, 32);
  if (lane == 0) red[wid] = lacc;
  __syncthreads();
  if (tid == 0) {
    float s = 0.f;
#pragma unroll
    for (int i = 0; i < 8; ++i) s += red[i];
    atomicAdd(loss, s);
  }
}

extern "C" void kernel_launch(void* const* d_in, const int* in_sizes, int n_in,
                              void* d_out, int out_size, void* d_ws, size_t ws_size,
                              hipStream_t stream) {
  (void)in_sizes; (void)n_in; (void)out_size; (void)d_ws; (void)ws_size;
  const float* outp = (const float*)d_in[0];
  const float* tgtp = (const float*)d_in[1];
  float* loss = (float*)d_out;
  gvl_zero<<<dim3(1), dim3(64), 0, stream>>>(loss);
  gvl_kernel<<<dim3(Wd / TILE, Hd / TILE, 16), dim3(256), 0, stream>>>(outp, tgtp, loss);
}